// RNNModel_65798898975039
// MI455X (gfx1250) — compile-verified
//
#include <hip/hip_runtime.h>
#include <hip/hip_bf16.h>
#include <math.h>

// ---------------------------------------------------------------------------
// Problem constants (from reference)
// ---------------------------------------------------------------------------
#define SEQ     70
#define BSZ     80
#define SB      (SEQ * BSZ)        // 5600
#define H       400                // NHID == NINP
#define NTILE_N 25                 // 400 / 16
#define NGROUP  5                  // groups of 5 N-tiles (80 columns)
#define NSAMP   50
#define NTOKEN  50000
#define TEMP    10.0f
#define REG_BIAS 1e-3f

typedef __attribute__((ext_vector_type(2))) float v2f;
typedef __attribute__((ext_vector_type(8))) float v8f;

// ---------------------------------------------------------------------------
// fp32 WMMA GEMM, 16x80 block per wave (5 N-tiles share one A stream):
//   out[m,n] = sum_k A[m,k] * W[n,k] + bias[n]
// A rows optionally gathered through a token index array (embedding lookup).
// grid.x = (M/16) * 5.
// ---------------------------------------------------------------------------
__global__ __launch_bounds__(32)
void gemm_wmma_f32(const float* __restrict__ Abase,
                   const int*   __restrict__ gather,   // nullptr -> direct rows
                   const float* __restrict__ W,        // [400,400] row-major
                   const float* __restrict__ bias,     // [400]
                   float*       __restrict__ out)      // [M,400]
{
    const int blk = blockIdx.x;
    const int m0 = (blk / NGROUP) * 16;
    const int g0 = (blk % NGROUP) * 80;      // first column of the 5-tile group

    const int lane = threadIdx.x;
    const int half = lane >> 4;              // K-pair {2h, 2h+1}
    const int l15  = lane & 15;

    const int m = m0 + l15;
    const float* arow = gather ? (Abase + (size_t)gather[m] * H)
                               : (Abase + (size_t)m * H);
    __builtin_prefetch(arow, 0, 1);

    const float* brow[NGROUP];
    #pragma unroll
    for (int t = 0; t < NGROUP; ++t)
        brow[t] = W + (size_t)(g0 + t * 16 + l15) * H;

    v8f acc[NGROUP];
    #pragma unroll
    for (int t = 0; t < NGROUP; ++t) acc[t] = (v8f){};

    #pragma unroll 2
    for (int k = 0; k < H; k += 4) {
        const int kb = k + half * 2;
        const v2f a = *(const v2f*)(arow + kb);
        #pragma unroll
        for (int t = 0; t < NGROUP; ++t) {
            const v2f b = *(const v2f*)(brow[t] + kb);
            acc[t] = __builtin_amdgcn_wmma_f32_16x16x4_f32(false, a, false, b,
                                                           (short)0, acc[t], false, false);
        }
    }

    #pragma unroll
    for (int t = 0; t < NGROUP; ++t) {
        const int col = g0 + t * 16 + l15;
        const float bn = bias[col];
        #pragma unroll
        for (int i = 0; i < 8; ++i) {
            const int row = m0 + i + 8 * half;
            out[(size_t)row * H + col] = acc[t][i] + bn;
        }
    }
}

// ---------------------------------------------------------------------------
// One RNN step: hnext = tanh(xW_t + hprev @ W_hh^T + b_hh)
// Latency-bound scan step: keep one tile per wave for max parallelism
// (M = 80 -> 5 tiles, N = 400 -> 25 tiles, 125 waves/step).
// ---------------------------------------------------------------------------
__global__ __launch_bounds__(32)
void rnn_step_wmma(const float* __restrict__ hprev,   // [80,400]
                   const float* __restrict__ Whh,     // [400,400]
                   const float* __restrict__ bhh,     // [400]
                   const float* __restrict__ xWt,     // [80,400]
                   float*       __restrict__ hnext)   // [80,400]
{
    const int tile = blockIdx.x;
    const int m0 = (tile / NTILE_N) * 16;
    const int n0 = (tile % NTILE_N) * 16;

    const int lane = threadIdx.x;
    const int half = lane >> 4;
    const int l15  = lane & 15;

    const float* arow = hprev + (size_t)(m0 + l15) * H;
    const float* brow = Whh   + (size_t)(n0 + l15) * H;

    v8f acc = {};
    #pragma unroll 4
    for (int k = 0; k < H; k += 4) {
        const int kb = k + half * 2;
        const v2f a = *(const v2f*)(arow + kb);
        const v2f b = *(const v2f*)(brow + kb);
        acc = __builtin_amdgcn_wmma_f32_16x16x4_f32(false, a, false, b,
                                                    (short)0, acc, false, false);
    }

    const int col = n0 + l15;
    const float bn = bhh[col];
    #pragma unroll
    for (int i = 0; i < 8; ++i) {
        const int row = m0 + i + 8 * half;
        hnext[(size_t)row * H + col] = tanhf(acc[i] + xWt[(size_t)row * H + col] + bn);
    }
}

// ---------------------------------------------------------------------------
// Negative samples (dominant GEMM), fused tanh + squared distance.
// One wave per strip of 16 positions of one sample:
//   strip -> n = strip/350, j0 = (strip%350)*16   (17500 strips)
// Inner: 5 groups of 5 N-tiles; within a group one A stream feeds 5 wmma
// chains (6 loads per 5 wmma). After each group, fuse
//   o = tanh(sW + b_ih + hU[j]),  dist[j] += (raw_flat[j]-o)^2
// Finally logits[1+n][j] = -TEMP*dist + dec_bias[samples[n,j]].
// ---------------------------------------------------------------------------
__global__ __launch_bounds__(32)
void neg_wmma(const int*   __restrict__ samples,   // [50, 5600]
              const float* __restrict__ emb,       // [50000, 400]
              const float* __restrict__ Wih,       // [400,400]
              const float* __restrict__ bih,       // [400]
              const float* __restrict__ hU,        // [5600,400]
              const float* __restrict__ rawflat,   // [5600,400]
              const float* __restrict__ dec_bias,  // [50000]
              float*       __restrict__ logits)    // [51, 5600]
{
    const int strip = blockIdx.x;
    const int n  = strip / (SB / 16);
    const int j0 = (strip % (SB / 16)) * 16;

    const int lane = threadIdx.x;
    const int half = lane >> 4;
    const int l15  = lane & 15;

    const int tok = samples[(size_t)n * SB + j0 + l15];
    const float* arow = emb + (size_t)tok * H;
    __builtin_prefetch(arow, 0, 1);

    float dist[8];
    #pragma unroll
    for (int i = 0; i < 8; ++i) dist[i] = 0.0f;

    for (int g = 0; g < NGROUP; ++g) {
        const int g0 = g * 80;

        const float* brow[NGROUP];
        #pragma unroll
        for (int t = 0; t < NGROUP; ++t)
            brow[t] = Wih + (size_t)(g0 + t * 16 + l15) * H;

        v8f acc[NGROUP];
        #pragma unroll
        for (int t = 0; t < NGROUP; ++t) acc[t] = (v8f){};

        #pragma unroll 2
        for (int k = 0; k < H; k += 4) {
            const int kb = k + half * 2;
            const v2f a = *(const v2f*)(arow + kb);
            #pragma unroll
            for (int t = 0; t < NGROUP; ++t) {
                const v2f b = *(const v2f*)(brow[t] + kb);
                acc[t] = __builtin_amdgcn_wmma_f32_16x16x4_f32(false, a, false, b,
                                                               (short)0, acc[t], false, false);
            }
        }

        #pragma unroll
        for (int t = 0; t < NGROUP; ++t) {
            const int col = g0 + t * 16 + l15;
            const float bn = bih[col];
            #pragma unroll
            for (int i = 0; i < 8; ++i) {
                const int j = j0 + i + 8 * half;
                const float sw = acc[t][i] + bn + hU[(size_t)j * H + col];
                const float o  = tanhf(sw);
                const float d  = rawflat[(size_t)j * H + col] - o;
                dist[i] += d * d;
            }
        }
    }

    // Reduce across the 16 lanes of each half (offsets < 16 stay in-half).
    #pragma unroll
    for (int i = 0; i < 8; ++i) {
        float v = dist[i];
        v += __shfl_xor(v, 1);
        v += __shfl_xor(v, 2);
        v += __shfl_xor(v, 4);
        v += __shfl_xor(v, 8);
        if (l15 == i) {
            const int j = j0 + i + 8 * half;
            const int tokj = samples[(size_t)n * SB + j];
            logits[(size_t)(1 + n) * SB + j] = -TEMP * v + dec_bias[tokj];
        }
    }
}

// ---------------------------------------------------------------------------
// Positive distances: logits[0][idx] = -TEMP*||raw[s,b]-raw[s+1,b]||^2 + dec_bias[data]
// One wave per output column.
// ---------------------------------------------------------------------------
__global__ __launch_bounds__(32)
void pos_kernel(const int* __restrict__ data, const float* __restrict__ raw,
                const float* __restrict__ dec_bias, float* __restrict__ logits)
{
    const int idx  = blockIdx.x;          // s*B + b
    const int lane = threadIdx.x;
    const float* p0 = raw + (size_t)idx * H;          // raw[s,b]
    const float* p1 = raw + (size_t)(idx + BSZ) * H;  // raw[s+1,b]
    float s = 0.0f;
    for (int k = lane; k < H; k += 32) {
        const float d = p0[k] - p1[k];
        s += d * d;
    }
    s += __shfl_xor(s, 1);
    s += __shfl_xor(s, 2);
    s += __shfl_xor(s, 4);
    s += __shfl_xor(s, 8);
    s += __shfl_xor(s, 16);
    if (lane == 0) logits[idx] = -TEMP * s + dec_bias[data[idx]];
}

// ---------------------------------------------------------------------------
// reg = REG_BIAS * sum(dec_bias^2)  (single block)
// ---------------------------------------------------------------------------
__global__ void reg_kernel(const float* __restrict__ dec_bias, float* __restrict__ regv)
{
    __shared__ float sbuf[256];
    const int tid = threadIdx.x;
    float s = 0.0f;
    for (int i = tid; i < NTOKEN; i += 256) {
        const float v = dec_bias[i];
        s += v * v;
    }
    sbuf[tid] = s;
    __syncthreads();
    for (int off = 128; off > 0; off >>= 1) {
        if (tid < off) sbuf[tid] += sbuf[tid + off];
        __syncthreads();
    }
    if (tid == 0) regv[0] = REG_BIAS * sbuf[0];
}

// ---------------------------------------------------------------------------
// Column log-softmax over 51 rows + reg scalar -> loss output
// ---------------------------------------------------------------------------
__global__ void softmax_kernel(const float* __restrict__ logits,
                               const float* __restrict__ regv,
                               float* __restrict__ out)
{
    const int j = blockIdx.x * blockDim.x + threadIdx.x;
    if (j >= SB) return;
    float mx = -INFINITY;
    for (int i = 0; i <= NSAMP; ++i) mx = fmaxf(mx, logits[(size_t)i * SB + j]);
    float se = 0.0f;
    for (int i = 0; i <= NSAMP; ++i) se += expf(logits[(size_t)i * SB + j] - mx);
    const float lse = mx + logf(se);
    const float reg = regv[0];
    for (int i = 0; i <= NSAMP; ++i)
        out[(size_t)i * SB + j] = logits[(size_t)i * SB + j] - lse + reg;
}

// ---------------------------------------------------------------------------
// Simple float copy
// ---------------------------------------------------------------------------
__global__ void copy_kernel(const float* __restrict__ src, float* __restrict__ dst, int n)
{
    const int i = blockIdx.x * blockDim.x + threadIdx.x;
    if (i < n) dst[i] = src[i];
}

// ---------------------------------------------------------------------------
// Host-side orchestration
// ---------------------------------------------------------------------------
extern "C" void kernel_launch(void* const* d_in, const int* in_sizes, int n_in,
                              void* d_out, int out_size, void* d_ws, size_t ws_size,
                              hipStream_t stream) {
    const int*   data     = (const int*)  d_in[0];   // [70,80]
    const int*   samples  = (const int*)  d_in[1];   // [50,5600]
    const float* hidden   = (const float*)d_in[2];   // [1,80,400]
    const float* emb      = (const float*)d_in[3];   // [50000,400]
    const float* W_ih     = (const float*)d_in[4];   // [400,400]
    const float* b_ih     = (const float*)d_in[5];   // [400]
    const float* W_hh     = (const float*)d_in[6];   // [400,400]
    const float* b_hh     = (const float*)d_in[7];   // [400]
    const float* dec_bias = (const float*)d_in[8];   // [50000]
    float* out = (float*)d_out;                      // 51*5600 loss + 80*400 hidden

    // Workspace layout (floats)
    float* ws     = (float*)d_ws;
    float* xW     = ws;                     // [5600,400]      2,240,000
    float* raw    = ws + 2240000;           // [71*80,400]     2,272,000
    float* hU     = ws + 4512000;           // [5600,400]      2,240,000
    float* logits = ws + 6752000;           // [51,5600]         285,600
    float* regv   = ws + 7037600;           // [1]

    // raw[0] = initial hidden (zeros per reference, but copy to be faithful)
    copy_kernel<<<(BSZ * H + 255) / 256, 256, 0, stream>>>(hidden, raw, BSZ * H);

    // xW = emb_table[data] @ W_ih^T + b_ih   (gathered GEMM, 350*5 blocks)
    gemm_wmma_f32<<<(SB / 16) * NGROUP, 32, 0, stream>>>(emb, data, W_ih, b_ih, xW);

    // 70-step sequential scan (dependency-ordered launches on the stream)
    for (int t = 0; t < SEQ; ++t) {
        rnn_step_wmma<<<(BSZ / 16) * NTILE_N, 32, 0, stream>>>(
            raw + (size_t)t * BSZ * H, W_hh, b_hh,
            xW  + (size_t)t * BSZ * H,
            raw + (size_t)(t + 1) * BSZ * H);
    }

    // hU = raw[:-1] @ W_hh^T + b_hh   (first 5600 rows of raw)
    gemm_wmma_f32<<<(SB / 16) * NGROUP, 32, 0, stream>>>(raw, nullptr, W_hh, b_hh, hU);

    // Negative samples: 17500 strips, fused GEMM + tanh + distance
    neg_wmma<<<NSAMP * (SB / 16), 32, 0, stream>>>(samples, emb, W_ih, b_ih,
                                                   hU, raw, dec_bias, logits);

    // Positive row
    pos_kernel<<<SB, 32, 0, stream>>>(data, raw, dec_bias, logits);

    // Regularizer scalar
    reg_kernel<<<1, 256, 0, stream>>>(dec_bias, regv);

    // Column log-softmax -> loss output
    softmax_kernel<<<(SB + 255) / 256, 256, 0, stream>>>(logits, regv, out);

    // new_hidden = raw[-1]  (rows 5600..5679)
    copy_kernel<<<(BSZ * H + 255) / 256, 256, 0, stream>>>(
        raw + (size_t)SB * H, out + (size_t)(NSAMP + 1) * SB, BSZ * H);
}